// DropoutPredictionGNN_67791763800740
// MI455X (gfx1250) — compile-verified
//
#include <hip/hip_runtime.h>
#include <hip/hip_bf16.h>
#include <math.h>

typedef __attribute__((ext_vector_type(2))) float v2f;
typedef __attribute__((ext_vector_type(8))) float v8f;

#define HIDW 128   // feature width (IN_DIM == HID == 128)

// ---------------------------------------------------------------------------
// Degree / normalization kernels
// ---------------------------------------------------------------------------
__global__ void deg_init_kernel(float* __restrict__ deg, int n) {
    int i = blockIdx.x * blockDim.x + threadIdx.x;
    if (i < n) deg[i] = 1.0f;   // self-loop contributes 1
}

__global__ void deg_count_kernel(const int* __restrict__ dst, float* __restrict__ deg, int nE) {
    int e = blockIdx.x * blockDim.x + threadIdx.x;
    if (e < nE) atomicAdd(&deg[dst[e]], 1.0f);
}

__global__ void dinv_kernel(float* __restrict__ deg, int n) {
    int i = blockIdx.x * blockDim.x + threadIdx.x;
    if (i < n) deg[i] = rsqrtf(deg[i]);   // in-place: deg -> d^-1/2
}

__global__ void zero_pool_kernel(float* __restrict__ pooled, float* __restrict__ cnts) {
    int i = blockIdx.x * blockDim.x + threadIdx.x;
    if (i < 64 * HIDW) pooled[i] = 0.0f;
    if (i < 64)        cnts[i]   = 0.0f;
}

// ---------------------------------------------------------------------------
// WMMA f32 GEMM:  H[M x 128] = A[M x 128] * W[128 x 128]
// Each wave owns one 16-wide column panel of W (all K=128, held in 64 VGPRs)
// and sweeps 4 consecutive 16-row M-tiles, 32x V_WMMA_F32_16X16X4_F32 each.
// A-frag (16x4, 32-bit): lanes 0-15 hold M=lane, K={k0,k0+1}; lanes 16-31 K={k0+2,k0+3}.
// B-frag (4x16): VGPR r holds row K=k0+r (lanes 0-15) / K=k0+r+2 (lanes 16-31), N=lane&15.
// C/D (16x16): VGPR r = row (r + (lane<16?0:8)), N=lane&15.
// ---------------------------------------------------------------------------
#define MT_PER_WAVE 4

__global__ __launch_bounds__(256)
void gemm_wmma_f32_kernel(const float* __restrict__ A, const float* __restrict__ W,
                          float* __restrict__ H, int M) {
    const int lane   = threadIdx.x & 31;
    const int wave   = threadIdx.x >> 5;
    const int ntile  = blockIdx.y;                      // 16-col tile index
    const int mtiles = (M + 15) >> 4;
    const int mbase  = (blockIdx.x * 8 + wave) * MT_PER_WAVE;
    if (mbase >= mtiles) return;                        // wave-uniform

    const int col   = ntile * 16 + (lane & 15);
    const int khalf = (lane < 16) ? 0 : 2;

    // Hoist the full-K B panel into registers (reused across MT_PER_WAVE tiles)
    v2f bfrag[32];
#pragma unroll
    for (int kk = 0; kk < 32; ++kk) {
        const float* wp = W + (size_t)(kk * 4 + khalf) * HIDW + col;
        bfrag[kk].x = wp[0];
        bfrag[kk].y = wp[HIDW];
    }

#pragma unroll
    for (int t = 0; t < MT_PER_WAVE; ++t) {
        const int mtile = mbase + t;
        if (mtile < mtiles) {                           // wave-uniform guard
            const float* __restrict__ arow =
                A + (size_t)(mtile * 16 + (lane & 15)) * HIDW + khalf;
            v8f acc = {};
#pragma unroll
            for (int kk = 0; kk < 32; ++kk) {
                v2f a;
                a.x = arow[kk * 4 + 0];
                a.y = arow[kk * 4 + 1];
                acc = __builtin_amdgcn_wmma_f32_16x16x4_f32(
                    /*neg_a=*/false, a, /*neg_b=*/false, bfrag[kk],
                    /*c_mod=*/(short)0, acc, /*reuse_a=*/false, /*reuse_b=*/false);
            }
            const int rbase = mtile * 16 + ((lane < 16) ? 0 : 8);
#pragma unroll
            for (int r = 0; r < 8; ++r)
                H[(size_t)(rbase + r) * HIDW + col] = acc[r];
        }
    }
}

// ---------------------------------------------------------------------------
// out[i,:] = H[i,:] * dinv[i]^2    (self-loop term; fully initializes `out`)
// One wave per node, float4 per lane (32 lanes * 4 = 128).
// ---------------------------------------------------------------------------
__global__ __launch_bounds__(256)
void selfloop_init_kernel(const float* __restrict__ H, const float* __restrict__ dinv,
                          float* __restrict__ out, int n) {
    int i = blockIdx.x * 8 + (threadIdx.x >> 5);
    if (i >= n) return;
    int lane = threadIdx.x & 31;
    float w = dinv[i] * dinv[i];
    float4 v = ((const float4*)(H + (size_t)i * HIDW))[lane];
    ((float4*)(out + (size_t)i * HIDW))[lane] =
        make_float4(v.x * w, v.y * w, v.z * w, v.w * w);
}

// ---------------------------------------------------------------------------
// Edge scatter: out[dst,:] += H[src,:] * dinv[src]*dinv[dst]
// One wave per edge; float4 load, 4 global f32 atomics per lane.
// ---------------------------------------------------------------------------
__global__ __launch_bounds__(256)
void edge_scatter_kernel(const float* __restrict__ H, const float* __restrict__ dinv,
                         const int* __restrict__ src, const int* __restrict__ dst,
                         float* __restrict__ out, int nE) {
    int e = blockIdx.x * 8 + (threadIdx.x >> 5);
    if (e >= nE) return;
    int lane = threadIdx.x & 31;
    int s = src[e];
    int d = dst[e];
    float w = dinv[s] * dinv[d];
    float4 v = ((const float4*)(H + (size_t)s * HIDW))[lane];
    float* od = out + (size_t)d * HIDW + lane * 4;
    atomicAdd(od + 0, v.x * w);
    atomicAdd(od + 1, v.y * w);
    atomicAdd(od + 2, v.z * w);
    atomicAdd(od + 3, v.w * w);
}

// ---------------------------------------------------------------------------
// out[i,f] += b[f]; optional ReLU
// ---------------------------------------------------------------------------
__global__ __launch_bounds__(256)
void bias_act_kernel(float* __restrict__ out, const float* __restrict__ b,
                     int total, int relu) {
    int idx = blockIdx.x * blockDim.x + threadIdx.x;
    if (idx >= total) return;
    float v = out[idx] + b[idx & (HIDW - 1)];
    out[idx] = relu ? fmaxf(v, 0.0f) : v;
}

// ---------------------------------------------------------------------------
// Pooling: pooled[batch[i],:] += h[i,:]; cnts[batch[i]] += 1
// ---------------------------------------------------------------------------
__global__ __launch_bounds__(256)
void pool_sum_kernel(const float* __restrict__ h, const int* __restrict__ batch,
                     float* __restrict__ pooled, float* __restrict__ cnts, int n) {
    int i = blockIdx.x * 8 + (threadIdx.x >> 5);
    if (i >= n) return;
    int lane = threadIdx.x & 31;
    int g = batch[i];
    float4 v = ((const float4*)(h + (size_t)i * HIDW))[lane];
    float* pg = pooled + (size_t)g * HIDW + lane * 4;
    atomicAdd(pg + 0, v.x);
    atomicAdd(pg + 1, v.y);
    atomicAdd(pg + 2, v.z);
    atomicAdd(pg + 3, v.w);
    if (lane == 0) atomicAdd(&cnts[g], 1.0f);
}

// ---------------------------------------------------------------------------
// Classifier MLP: sigmoid(relu(mean_pool @ Wc1 + bc1) @ Wc2 + bc2)
// One block (64 threads) per graph; thread j owns hidden unit j.
// ---------------------------------------------------------------------------
__global__ __launch_bounds__(64)
void classifier_kernel(const float* __restrict__ pooled, const float* __restrict__ cnts,
                       const float* __restrict__ Wc1, const float* __restrict__ bc1,
                       const float* __restrict__ Wc2, const float* __restrict__ bc2,
                       float* __restrict__ out) {
    int g = blockIdx.x;
    int j = threadIdx.x;   // 0..63
    __shared__ float red[64];
    float inv = 1.0f / fmaxf(cnts[g], 1.0f);
    float z = bc1[j];
    const float* pg = pooled + (size_t)g * HIDW;
#pragma unroll 4
    for (int k = 0; k < HIDW; ++k)
        z += pg[k] * inv * Wc1[(size_t)k * 64 + j];
    red[j] = fmaxf(z, 0.0f) * Wc2[j];
    __syncthreads();
    for (int s = 32; s > 0; s >>= 1) {
        if (j < s) red[j] += red[j + s];
        __syncthreads();
    }
    if (j == 0) out[g] = 1.0f / (1.0f + expf(-(red[0] + bc2[0])));
}

// ---------------------------------------------------------------------------
// Host launch
// ---------------------------------------------------------------------------
extern "C" void kernel_launch(void* const* d_in, const int* in_sizes, int n_in,
                              void* d_out, int out_size, void* d_ws, size_t ws_size,
                              hipStream_t stream) {
    const float* x    = (const float*)d_in[0];
    const int*   ei   = (const int*)d_in[1];
    const int*   bat  = (const int*)d_in[2];
    const float* W1   = (const float*)d_in[3];
    const float* b1   = (const float*)d_in[4];
    const float* W2   = (const float*)d_in[5];
    const float* b2   = (const float*)d_in[6];
    const float* Wc1  = (const float*)d_in[7];
    const float* bc1  = (const float*)d_in[8];
    const float* Wc2  = (const float*)d_in[9];
    const float* bc2  = (const float*)d_in[10];
    float* out = (float*)d_out;

    const int n  = in_sizes[0] / HIDW;   // 50000 nodes
    const int nE = in_sizes[1] / 2;      // 800000 edges
    const int* src = ei;
    const int* dst = ei + nE;

    // Workspace layout
    float* bufH   = (float*)d_ws;                       // [n,128] pre-aggregation features
    float* bufA   = bufH + (size_t)n * HIDW;            // [n,128] aggregated features
    float* dinv   = bufA + (size_t)n * HIDW;            // [n]  deg -> d^-1/2
    float* pooled = dinv + ((n + 63) & ~63);            // [64,128]
    float* cnts   = pooled + 64 * HIDW;                 // [64]

    const int T = 256;
    const int mtiles = (n + 15) / 16;
    dim3 gemmGrid((mtiles + 8 * MT_PER_WAVE - 1) / (8 * MT_PER_WAVE), HIDW / 16);
    int nodeWaveBlocks = (n + 7) / 8;
    int edgeWaveBlocks = (nE + 7) / 8;

    // --- degree / normalization ---
    deg_init_kernel<<<(n + T - 1) / T, T, 0, stream>>>(dinv, n);
    deg_count_kernel<<<(nE + T - 1) / T, T, 0, stream>>>(dst, dinv, nE);
    dinv_kernel<<<(n + T - 1) / T, T, 0, stream>>>(dinv, n);
    zero_pool_kernel<<<(64 * HIDW + T - 1) / T, T, 0, stream>>>(pooled, cnts);

    // --- layer 1 ---
    gemm_wmma_f32_kernel<<<gemmGrid, T, 0, stream>>>(x, W1, bufH, n);
    selfloop_init_kernel<<<nodeWaveBlocks, T, 0, stream>>>(bufH, dinv, bufA, n);
    edge_scatter_kernel<<<edgeWaveBlocks, T, 0, stream>>>(bufH, dinv, src, dst, bufA, nE);
    bias_act_kernel<<<((size_t)n * HIDW + T - 1) / T, T, 0, stream>>>(bufA, b1, n * HIDW, 1);

    // --- layer 2 (bufA -> bufH -> bufA) ---
    gemm_wmma_f32_kernel<<<gemmGrid, T, 0, stream>>>(bufA, W2, bufH, n);
    selfloop_init_kernel<<<nodeWaveBlocks, T, 0, stream>>>(bufH, dinv, bufA, n);
    edge_scatter_kernel<<<edgeWaveBlocks, T, 0, stream>>>(bufH, dinv, src, dst, bufA, nE);
    bias_act_kernel<<<((size_t)n * HIDW + T - 1) / T, T, 0, stream>>>(bufA, b2, n * HIDW, 0);

    // --- pool + classifier ---
    pool_sum_kernel<<<nodeWaveBlocks, T, 0, stream>>>(bufA, bat, pooled, cnts, n);
    classifier_kernel<<<64, 64, 0, stream>>>(pooled, cnts, Wc1, bc1, Wc2, bc2, out);

    (void)n_in; (void)out_size; (void)ws_size;
}